// SequencePairAttention_54211077210608
// MI455X (gfx1250) — compile-verified
//
#include <hip/hip_runtime.h>

// SequencePairAttention on MI455X (gfx1250), wave32, WMMA f32 16x16x4.
//
// Shapes: N=4, L=512, E=256, PD=64, H=8, HD=32.
// Bandwidth-bound problem (~6 GFLOP vs ~0.85 GB HBM floor) -> keep f32,
// use V_WMMA_F32_16X16X4_F32 for every GEMM.
// Cache policy: pair_rep (268 MB, streamed twice) and pair_out2 (268 MB,
// written once) use non-temporal hints so the 32 MB logits buffer (touched
// 5x) stays resident in the 192 MB L2.

#define NB  4
#define LL  512
#define EE  256
#define PDD 64
#define HH  8
#define HDD 32
#define HPD 512   // H*PD
#define ROWS 2048 // N*L

typedef __attribute__((ext_vector_type(2))) float v2f;
typedef __attribute__((ext_vector_type(4))) float v4f;
typedef __attribute__((ext_vector_type(8))) float v8f;

// D = A(16x4) * B(4x16) + C, f32, wave32.
// A layout: lanes 0-15 hold rows 0-15 with K={k0,k0+1} in vgpr{0,1};
//           lanes 16-31 hold rows 0-15 with K={k0+2,k0+3}.
// B mirrored with columns across lanes. C/D: vgpr r -> M=r (lanes 0-15),
// M=8+r (lanes 16-31), N = lane&15.
__device__ __forceinline__ v8f wmma_f32x4(v2f a, v2f b, v8f c) {
  return __builtin_amdgcn_wmma_f32_16x16x4_f32(
      /*neg_a=*/false, a, /*neg_b=*/false, b,
      /*c_mod=*/(short)0, c, /*reuse_a=*/false, /*reuse_b=*/false);
}

// ---------------------------------------------------------------------------
// K1: q/k/v projections.  [2048 x 256] @ [256 x 256] + bias, one of q/k/v per
// blockIdx.z.  8 waves/block, each wave owns one 16x16 output tile; the 16x256
// A tile is staged in LDS and shared by all 8 waves.
// ---------------------------------------------------------------------------
__global__ void k_qkv(const float* __restrict__ X,
                      const float* __restrict__ W0, const float* __restrict__ B0,
                      const float* __restrict__ W1, const float* __restrict__ B1,
                      const float* __restrict__ W2, const float* __restrict__ B2,
                      float* __restrict__ O0, float* __restrict__ O1,
                      float* __restrict__ O2) {
  __shared__ float As[16 * EE];
  const int tid = threadIdx.x;
  const int rt  = blockIdx.x;          // 128 row tiles
  const int which = blockIdx.z;
  const float* W  = (which == 0) ? W0 : (which == 1) ? W1 : W2;
  const float* Bi = (which == 0) ? B0 : (which == 1) ? B1 : B2;
  float*       O  = (which == 0) ? O0 : (which == 1) ? O1 : O2;

  for (int i = tid; i < 16 * (EE / 4); i += 256) {
    int r = i >> 6, c4 = i & 63;
    ((float4*)As)[i] = ((const float4*)(X + (size_t)(rt * 16 + r) * EE))[c4];
  }
  __syncthreads();

  const int wave = tid >> 5, lane = tid & 31;
  const int half = lane >> 4, lr = lane & 15;
  const int col  = blockIdx.y * 128 + wave * 16 + lr;

  v8f acc = {};
  for (int k0 = 0; k0 < EE; k0 += 4) {
    const int ka = k0 + 2 * half;
    if (k0 + 8 < EE) __builtin_prefetch(&W[(size_t)(ka + 8) * EE + col], 0, 1);
    v2f a, b;
    a.x = As[lr * EE + ka];
    a.y = As[lr * EE + ka + 1];
    b.x = W[(size_t)ka * EE + col];
    b.y = W[(size_t)(ka + 1) * EE + col];
    acc = wmma_f32x4(a, b, acc);
  }
  const float bias = Bi[col];
#pragma unroll
  for (int r = 0; r < 8; ++r) {
    const int row = rt * 16 + half * 8 + r;
    O[(size_t)row * EE + col] = acc[r] + bias;
  }
}

// ---------------------------------------------------------------------------
// K2a: dpa logits.  Per (n,h): [512 x 32] @ [32 x 512] -> logits[n,h,q,v].
// ---------------------------------------------------------------------------
__global__ void k_dpa(const float* __restrict__ Q, const float* __restrict__ Kb,
                      float* __restrict__ logits) {
  __shared__ float As[16 * HDD];
  const int tid = threadIdx.x;
  const int qt  = blockIdx.x;      // 32 q tiles
  const int nh  = blockIdx.z;      // n*8+h
  const int n = nh >> 3, h = nh & 7;

  for (int i = tid; i < 16 * (HDD / 4); i += 256) {   // 128 float4
    int r = i >> 3, c4 = i & 7;
    ((float4*)As)[i] =
        ((const float4*)(Q + (((size_t)n * LL + qt * 16 + r) * HH + h) * HDD))[c4];
  }
  __syncthreads();

  const int wave = tid >> 5, lane = tid & 31;
  const int half = lane >> 4, lr = lane & 15;
  const int vcol = blockIdx.y * 128 + wave * 16 + lr;

  v8f acc = {};
  for (int k0 = 0; k0 < HDD; k0 += 4) {
    const int ka = k0 + 2 * half;
    v2f a, b;
    a.x = As[lr * HDD + ka];
    a.y = As[lr * HDD + ka + 1];
    b.x = Kb[(((size_t)n * LL + vcol) * HH + h) * HDD + ka];
    b.y = Kb[(((size_t)n * LL + vcol) * HH + h) * HDD + ka + 1];
    acc = wmma_f32x4(a, b, acc);
  }
  float* out = logits + ((size_t)nh * LL + qt * 16) * LL;
#pragma unroll
  for (int r = 0; r < 8; ++r) {
    const int qrow = half * 8 + r;
    out[(size_t)qrow * LL + vcol] = acc[r];
  }
}

// ---------------------------------------------------------------------------
// K2b: logits[n,h,q,v] += pair_rep[n,q,v,:] @ Wpt[:,h] + bpt[h].
// Streaming pass 1 over pair_rep: non-temporal b128 loads (stream-once,
// keep L2 for logits); coalesced RMW into logits (L2-resident).
// ---------------------------------------------------------------------------
__global__ void k_pair_bias(const float* __restrict__ PR,
                            const float* __restrict__ Wpt,
                            const float* __restrict__ bpt,
                            float* __restrict__ logits) {
  __shared__ float Ws[PDD * HH];
  __shared__ float Bs[HH];
  const int tid = threadIdx.x;
  for (int i = tid; i < PDD * HH; i += 256) Ws[i] = Wpt[i];
  if (tid < HH) Bs[tid] = bpt[tid];
  __syncthreads();

  const int gid = blockIdx.x * 256 + tid;     // n*L*L + q*L + v
  const int n  = gid >> 18;
  const int qq = (gid >> 9) & (LL - 1);
  const int vv = gid & (LL - 1);
  const v4f* pr4 = (const v4f*)(PR + (size_t)gid * PDD);

  float acc[HH];
#pragma unroll
  for (int hh = 0; hh < HH; ++hh) acc[hh] = Bs[hh];
#pragma unroll
  for (int c4 = 0; c4 < PDD / 4; ++c4) {
    const v4f x = __builtin_nontemporal_load(pr4 + c4);   // th:NT
#pragma unroll
    for (int j = 0; j < 4; ++j)
#pragma unroll
      for (int hh = 0; hh < HH; ++hh)
        acc[hh] += x[j] * Ws[(c4 * 4 + j) * HH + hh];
  }
#pragma unroll
  for (int hh = 0; hh < HH; ++hh) {
    const size_t o = (((size_t)(n * HH + hh) * LL + qq) * LL + vv);
    logits[o] += acc[hh];
  }
}

// ---------------------------------------------------------------------------
// K3: softmax over v.  One wave32 per row of 512; 16 elems/lane in registers,
// max/sum via __shfl_xor butterfly (wave32!).
// ---------------------------------------------------------------------------
__global__ void k_softmax(float* __restrict__ logits) {
  const int tid  = threadIdx.x;
  const int wave = tid >> 5, lane = tid & 31;
  const size_t row = (size_t)blockIdx.x * 8 + wave;    // < N*H*L = 16384
  float* p = logits + row * LL;

  float vals[16];
  float m = -3.4e38f;
#pragma unroll
  for (int i = 0; i < 16; ++i) {
    vals[i] = p[lane + i * 32];
    m = fmaxf(m, vals[i]);
  }
#pragma unroll
  for (int off = 16; off >= 1; off >>= 1) m = fmaxf(m, __shfl_xor(m, off, 32));
  float s = 0.f;
#pragma unroll
  for (int i = 0; i < 16; ++i) {
    vals[i] = __expf(vals[i] - m);
    s += vals[i];
  }
#pragma unroll
  for (int off = 16; off >= 1; off >>= 1) s += __shfl_xor(s, off, 32);
  const float inv = 1.0f / s;
#pragma unroll
  for (int i = 0; i < 16; ++i) p[lane + i * 32] = vals[i] * inv;
}

// ---------------------------------------------------------------------------
// K4: attn_out[n,q,h,c] = sum_v aw[n,h,q,v] * v[n,v,h,c].
// Per (n,h): [512 x 512] @ [512 x 32]; 2 waves/block (one per 16-col tile),
// 16x512 aw tile staged in LDS.
// ---------------------------------------------------------------------------
__global__ void k_attn_v(const float* __restrict__ AW, const float* __restrict__ V,
                         float* __restrict__ AO) {
  __shared__ float As[16 * LL];                // 32 KB
  const int tid = threadIdx.x;                 // 64 threads
  const int qt  = blockIdx.x;
  const int nh  = blockIdx.y;
  const int n = nh >> 3, h = nh & 7;
  const float* src = AW + ((size_t)nh * LL + qt * 16) * LL;
  for (int i = tid; i < 16 * (LL / 4); i += 64) {
    int r = i >> 7, c4 = i & 127;
    ((float4*)As)[i] = ((const float4*)(src + (size_t)r * LL))[c4];
  }
  __syncthreads();

  const int wave = tid >> 5, lane = tid & 31;  // wave == col tile (0/1)
  const int half = lane >> 4, lr = lane & 15;
  const int ccol = wave * 16 + lr;             // 0..31

  v8f acc = {};
  for (int k0 = 0; k0 < LL; k0 += 4) {
    const int ka = k0 + 2 * half;
    v2f a, b;
    a.x = As[lr * LL + ka];
    a.y = As[lr * LL + ka + 1];
    b.x = V[(((size_t)n * LL + ka) * HH + h) * HDD + ccol];
    b.y = V[(((size_t)n * LL + ka + 1) * HH + h) * HDD + ccol];
    acc = wmma_f32x4(a, b, acc);
  }
#pragma unroll
  for (int r = 0; r < 8; ++r) {
    const int qrow = qt * 16 + half * 8 + r;
    AO[((size_t)n * LL + qrow) * EE + h * HDD + ccol] = acc[r];
  }
}

// ---------------------------------------------------------------------------
// K5: pair_out[n,q,h,c] = sum_v aw[n,h,q,v] * pair_rep[n,v,q,c].
// Per (n,q): [8(pad16) x 512] @ [512 x 64]; 4 waves/block, one col tile each.
// Streaming pass 2 over pair_rep: non-temporal B loads.
// ---------------------------------------------------------------------------
__global__ void k_pair_out(const float* __restrict__ AW,
                           const float* __restrict__ PR,
                           float* __restrict__ PO) {
  __shared__ float As[HH * LL];                // 8x512 = 16 KB
  const int tid = threadIdx.x;                 // 128 threads
  const int nq  = blockIdx.x;                  // n*512+q
  const int n = nq >> 9, qq = nq & (LL - 1);
  for (int i = tid; i < HH * (LL / 4); i += 128) {
    int r = i >> 7, c4 = i & 127;
    ((float4*)As)[i] =
        ((const float4*)(AW + ((size_t)(n * HH + r) * LL + qq) * LL))[c4];
  }
  __syncthreads();

  const int wave = tid >> 5, lane = tid & 31;  // wave == col tile (0..3)
  const int half = lane >> 4, lr = lane & 15;
  const int ccol = wave * 16 + lr;             // 0..63

  v8f acc = {};
  for (int k0 = 0; k0 < LL; k0 += 4) {
    const int ka = k0 + 2 * half;
    v2f a, b;
    a.x = (lr < HH) ? As[lr * LL + ka] : 0.f;      // pad rows 8..15 with 0
    a.y = (lr < HH) ? As[lr * LL + ka + 1] : 0.f;
    b.x = __builtin_nontemporal_load(
        &PR[(((size_t)n * LL + ka) * LL + qq) * PDD + ccol]);
    b.y = __builtin_nontemporal_load(
        &PR[(((size_t)n * LL + ka + 1) * LL + qq) * PDD + ccol]);
    acc = wmma_f32x4(a, b, acc);
  }
  if (half == 0) {                              // rows 0..7 == valid heads
#pragma unroll
    for (int r = 0; r < 8; ++r)
      PO[(size_t)nq * HPD + r * PDD + ccol] = acc[r];
  }
}

// ---------------------------------------------------------------------------
// K6: out = attn_out @ Wvt + bvt + pair_out @ Wpe + bpe   [2048 x 256]
// Two K-loops accumulate into the same WMMA accumulator.
// ---------------------------------------------------------------------------
__global__ void k_out_final(const float* __restrict__ AO,
                            const float* __restrict__ PO,
                            const float* __restrict__ Wvt,
                            const float* __restrict__ bvt,
                            const float* __restrict__ Wpe,
                            const float* __restrict__ bpe,
                            float* __restrict__ OUT) {
  __shared__ float AsA[16 * EE];    // 16 KB
  __shared__ float AsP[16 * HPD];   // 32 KB
  const int tid = threadIdx.x;
  const int rt  = blockIdx.x;
  for (int i = tid; i < 16 * (EE / 4); i += 256) {
    int r = i >> 6, c4 = i & 63;
    ((float4*)AsA)[i] = ((const float4*)(AO + (size_t)(rt * 16 + r) * EE))[c4];
  }
  for (int i = tid; i < 16 * (HPD / 4); i += 256) {
    int r = i >> 7, c4 = i & 127;
    ((float4*)AsP)[i] = ((const float4*)(PO + (size_t)(rt * 16 + r) * HPD))[c4];
  }
  __syncthreads();

  const int wave = tid >> 5, lane = tid & 31;
  const int half = lane >> 4, lr = lane & 15;
  const int col  = blockIdx.y * 128 + wave * 16 + lr;

  v8f acc = {};
  for (int k0 = 0; k0 < EE; k0 += 4) {
    const int ka = k0 + 2 * half;
    v2f a, b;
    a.x = AsA[lr * EE + ka];
    a.y = AsA[lr * EE + ka + 1];
    b.x = Wvt[(size_t)ka * EE + col];
    b.y = Wvt[(size_t)(ka + 1) * EE + col];
    acc = wmma_f32x4(a, b, acc);
  }
  for (int k0 = 0; k0 < HPD; k0 += 4) {
    const int ka = k0 + 2 * half;
    v2f a, b;
    a.x = AsP[lr * HPD + ka];
    a.y = AsP[lr * HPD + ka + 1];
    b.x = Wpe[(size_t)ka * EE + col];
    b.y = Wpe[(size_t)(ka + 1) * EE + col];
    acc = wmma_f32x4(a, b, acc);
  }
  const float bias = bvt[col] + bpe[col];
#pragma unroll
  for (int r = 0; r < 8; ++r) {
    const int row = rt * 16 + half * 8 + r;
    OUT[(size_t)row * EE + col] = acc[r] + bias;
  }
}

// ---------------------------------------------------------------------------
// K7: p = pair_out @ Wpp + bpp   [2048 x 64]
// ---------------------------------------------------------------------------
__global__ void k_p(const float* __restrict__ PO, const float* __restrict__ Wpp,
                    const float* __restrict__ bpp, float* __restrict__ P) {
  __shared__ float As[16 * HPD];    // 32 KB
  const int tid = threadIdx.x;      // 128 threads
  const int rt  = blockIdx.x;
  for (int i = tid; i < 16 * (HPD / 4); i += 128) {
    int r = i >> 7, c4 = i & 127;
    ((float4*)As)[i] = ((const float4*)(PO + (size_t)(rt * 16 + r) * HPD))[c4];
  }
  __syncthreads();

  const int wave = tid >> 5, lane = tid & 31;  // wave == col tile (0..3)
  const int half = lane >> 4, lr = lane & 15;
  const int col  = wave * 16 + lr;             // 0..63

  v8f acc = {};
  for (int k0 = 0; k0 < HPD; k0 += 4) {
    const int ka = k0 + 2 * half;
    v2f a, b;
    a.x = As[lr * HPD + ka];
    a.y = As[lr * HPD + ka + 1];
    b.x = Wpp[(size_t)ka * PDD + col];
    b.y = Wpp[(size_t)(ka + 1) * PDD + col];
    acc = wmma_f32x4(a, b, acc);
  }
  const float bias = bpp[col];
#pragma unroll
  for (int r = 0; r < 8; ++r) {
    const int row = rt * 16 + half * 8 + r;
    P[(size_t)row * PDD + col] = acc[r] + bias;
  }
}

// ---------------------------------------------------------------------------
// K8: pair_out2[n,i,j,z] = p[n,i,z] * p[n,j,z].  Pure streaming write of
// 268 MB: non-temporal b128 stores (written once, never re-read); p stays
// cached (read many times), p[n,i,:] broadcast from LDS.
// ---------------------------------------------------------------------------
__global__ void k_outer(const float* __restrict__ P, float* __restrict__ OUT2) {
  __shared__ float pi[PDD];
  const int tid = threadIdx.x;                 // 256 threads
  const int n = blockIdx.z, ii = blockIdx.y, jt = blockIdx.x;
  if (tid < PDD) pi[tid] = P[((size_t)n * LL + ii) * PDD + tid];
  __syncthreads();
  const int j  = jt * 16 + (tid >> 4);
  const int z4 = (tid & 15) * 4;
  const v4f pj = *((const v4f*)(P + ((size_t)n * LL + j) * PDD + z4));
  v4f o;
  o.x = pi[z4 + 0] * pj.x;
  o.y = pi[z4 + 1] * pj.y;
  o.z = pi[z4 + 2] * pj.z;
  o.w = pi[z4 + 3] * pj.w;
  __builtin_nontemporal_store(
      o, (v4f*)(OUT2 + (((size_t)n * LL + ii) * LL + j) * PDD + z4));  // th:NT
}

// ---------------------------------------------------------------------------
extern "C" void kernel_launch(void* const* d_in, const int* in_sizes, int n_in,
                              void* d_out, int out_size, void* d_ws,
                              size_t ws_size, hipStream_t stream) {
  const float* emb = (const float*)d_in[0];
  const float* prd = (const float*)d_in[1];
  const float* Wq  = (const float*)d_in[2];  const float* bq  = (const float*)d_in[3];
  const float* Wk  = (const float*)d_in[4];  const float* bk  = (const float*)d_in[5];
  const float* Wv  = (const float*)d_in[6];  const float* bv  = (const float*)d_in[7];
  const float* Wvt = (const float*)d_in[8];  const float* bvt = (const float*)d_in[9];
  const float* Wpt = (const float*)d_in[10]; const float* bpt = (const float*)d_in[11];
  const float* Wpe = (const float*)d_in[12]; const float* bpe = (const float*)d_in[13];
  const float* Wpp = (const float*)d_in[14]; const float* bpp = (const float*)d_in[15];

  float* out  = (float*)d_out;                       // [N,L,E]
  float* out2 = out + (size_t)NB * LL * EE;          // [N,L,L,PD]

  // Workspace layout (floats), total ~46.7 MB:
  float* ws = (float*)d_ws;
  float* qb     = ws;                                      // 2048*256
  float* kb     = qb + (size_t)ROWS * EE;                  // 2048*256
  float* vb     = kb + (size_t)ROWS * EE;                  // 2048*256
  float* logits = vb + (size_t)ROWS * EE;                  // 4*8*512*512 (reused as aw)
  float* ao     = logits + (size_t)NB * HH * LL * LL;      // 2048*256
  float* po     = ao + (size_t)ROWS * EE;                  // 2048*512
  float* pv     = po + (size_t)ROWS * HPD;                 // 2048*64

  k_qkv      <<<dim3(128, 2, 3),  256, 0, stream>>>(emb, Wq, bq, Wk, bk, Wv, bv, qb, kb, vb);
  k_dpa      <<<dim3(32, 4, 32),  256, 0, stream>>>(qb, kb, logits);
  k_pair_bias<<<dim3(4096),       256, 0, stream>>>(prd, Wpt, bpt, logits);
  k_softmax  <<<dim3(2048),       256, 0, stream>>>(logits);
  k_attn_v   <<<dim3(32, 32),      64, 0, stream>>>(logits, vb, ao);
  k_pair_out <<<dim3(2048),       128, 0, stream>>>(logits, prd, po);
  k_out_final<<<dim3(128, 2),     256, 0, stream>>>(ao, po, Wvt, bvt, Wpe, bpe, out);
  k_p        <<<dim3(128),        128, 0, stream>>>(po, Wpp, bpp, pv);
  k_outer    <<<dim3(32, 512, 4), 256, 0, stream>>>(pv, out2);
}